// Sbox16_82343112999380
// MI455X (gfx1250) — compile-verified
//
#include <hip/hip_runtime.h>
#include <hip/hip_bf16.h>

typedef __attribute__((ext_vector_type(2))) float v2f;
typedef __attribute__((ext_vector_type(8))) float v8f;

#define B_ROWS 32768
#define IN_DIM 128
#define HID    4096
#define SEGW   256
#define NSEG   16

// ---------------------------------------------------------------------------
// Kernel 0: fold W3 @ W2 into MT, stored transposed: MT[col*128 + j].
// One wave computes one 16x16 tile of M via 1024 chained f32 WMMAs (K=4096).
// ---------------------------------------------------------------------------
__global__ void fold_w3w2_kernel(const float* __restrict__ W2,
                                 const float* __restrict__ W3,
                                 float* __restrict__ MT) {
  const int wave = (blockIdx.x * blockDim.x + threadIdx.x) >> 5;
  const int lane = threadIdx.x & 31;
  const int jt = wave >> 8;          // 0..7   (128/16 row tiles of M)
  const int ct = wave & 255;         // 0..255 (4096/16 col tiles of M)
  const int j0 = jt * 16;
  const int c0 = ct * 16;
  const int m  = lane & 15;          // A-row / B-col within tile
  const int hi = lane >> 4;

  v8f acc = {};
  for (int k0 = 0; k0 < HID; k0 += 4) {
    const int ka = k0 + 2 * hi;
    // A[m][ka], A[m][ka+1] : W3 is [128, 4096] row-major -> contiguous in k
    v2f a = *(const v2f*)(W3 + (size_t)(j0 + m) * HID + ka);
    // B[ka][n], B[ka+1][n] : B[k][n] = W2[k][c0+n], W2 is [4096,4096] row-major
    v2f b;
    b.x = W2[(size_t)ka * HID + c0 + m];
    b.y = W2[(size_t)(ka + 1) * HID + c0 + m];
    acc = __builtin_amdgcn_wmma_f32_16x16x4_f32(false, a, false, b,
                                                (short)0, acc, false, false);
  }
#pragma unroll
  for (int r = 0; r < 8; ++r) {
    const int j = j0 + r + 8 * hi;
    MT[(size_t)(c0 + m) * IN_DIM + j] = acc[r];
  }
}

// ---------------------------------------------------------------------------
// Kernel 1: c[j] = b3[j] + dot(W3[j, :], b2)
// ---------------------------------------------------------------------------
__global__ void fold_bias_kernel(const float* __restrict__ W3,
                                 const float* __restrict__ b2,
                                 const float* __restrict__ b3,
                                 float* __restrict__ cvec) {
  const int j = threadIdx.x; // 128 threads
  float s = b3[j];
  for (int n = 0; n < HID; ++n) s += W3[(size_t)j * HID + n] * b2[n];
  cvec[j] = s;
}

// ---------------------------------------------------------------------------
// Main kernel: per 16-row tile
//   phase 1: o = x @ W1.T + b1 via WMMA f32 16x16x4, staged to LDS (256 KB)
//   phase 2: per-(row, segment) argmax over 256 (first-max semantics)
//   phase 3: out[row] = c + sum_k MT[col_k, :]   (coalesced over j)
// ---------------------------------------------------------------------------
__global__ void mlp_argmax_kernel(const float* __restrict__ x,
                                  const float* __restrict__ W1,
                                  const float* __restrict__ b1,
                                  const float* __restrict__ MT,
                                  const float* __restrict__ cvec,
                                  float* __restrict__ out) {
  extern __shared__ float smem[];
  float* o_lds  = smem;                       // 16 * 4096 floats
  int*   idx_lds = (int*)(smem + 16 * HID);   // 16 * 16 ints

  const int tid  = threadIdx.x;               // 256 threads = 8 waves
  const int wave = tid >> 5;
  const int lane = tid & 31;
  const int m    = lane & 15;
  const int hi   = lane >> 4;
  const int row0 = blockIdx.x * 16;

  // ---- phase 1: GEMM1 into LDS ----
  // Preload the wave's A tile (x rows) into registers: 32 k-steps of v2f.
  v2f a[32];
  {
    const float* xrow = x + (size_t)(row0 + m) * IN_DIM + 2 * hi;
#pragma unroll
    for (int kk = 0; kk < 32; ++kk) a[kk] = *(const v2f*)(xrow + kk * 4);
  }

  const int cbase = wave * 512;               // each wave owns 2 segments
  for (int t = 0; t < 32; ++t) {
    const int c0 = cbase + t * 16;
    const float bn = b1[c0 + m];              // bias depends on column only
    v8f acc = {bn, bn, bn, bn, bn, bn, bn, bn};
#pragma unroll
    for (int kk = 0; kk < 32; ++kk) {
      const int ka = kk * 4 + 2 * hi;
      // B[k][n] = W1[c0+n][k], W1 is [4096,128] row-major -> contiguous in k
      v2f b = *(const v2f*)(W1 + (size_t)(c0 + m) * IN_DIM + ka);
      acc = __builtin_amdgcn_wmma_f32_16x16x4_f32(false, a[kk], false, b,
                                                  (short)0, acc, false, false);
    }
#pragma unroll
    for (int r = 0; r < 8; ++r)
      o_lds[(r + 8 * hi) * HID + c0 + m] = acc[r];
  }
  __syncthreads();

  // ---- phase 2: argmax per (row, segment), first-index tie-break ----
  {
    const int row = tid >> 4;                 // 0..15
    const int seg = tid & 15;                 // 0..15
    const float* p = o_lds + row * HID + seg * SEGW;
    float best = p[0];
    int bi = 0;
    for (int j = 1; j < SEGW; ++j) {
      const float v = p[j];
      if (v > best) { best = v; bi = j; }
    }
    idx_lds[row * NSEG + seg] = seg * SEGW + bi;
  }
  __syncthreads();

  // ---- phase 3: gather-sum from MT (L2-resident, coalesced over j) ----
  {
    const int j    = tid & 127;
    const int rsel = tid >> 7;                // 0..1
#pragma unroll
    for (int p8 = 0; p8 < 8; ++p8) {
      const int row = rsel * 8 + p8;
      float s = cvec[j];
#pragma unroll
      for (int k = 0; k < NSEG; ++k) {
        const int col = idx_lds[row * NSEG + k];
        s += MT[(size_t)col * IN_DIM + j];
      }
      out[(size_t)(row0 + row) * IN_DIM + j] = s;
    }
  }
}

// ---------------------------------------------------------------------------
extern "C" void kernel_launch(void* const* d_in, const int* in_sizes, int n_in,
                              void* d_out, int out_size, void* d_ws, size_t ws_size,
                              hipStream_t stream) {
  const float* x  = (const float*)d_in[0];
  const float* W1 = (const float*)d_in[1];
  const float* b1 = (const float*)d_in[2];
  const float* W2 = (const float*)d_in[3];
  const float* b2 = (const float*)d_in[4];
  const float* W3 = (const float*)d_in[5];
  const float* b3 = (const float*)d_in[6];
  float* out = (float*)d_out;

  float* MT   = (float*)d_ws;                 // [4096, 128] = 2 MB
  float* cvec = MT + (size_t)HID * IN_DIM;    // [128]

  // Fold batch-independent weights: M = W3 @ W2 (stored transposed), c = W3@b2 + b3
  fold_w3w2_kernel<<<256, 256, 0, stream>>>(W2, W3, MT);   // 2048 waves
  fold_bias_kernel<<<1, 128, 0, stream>>>(W3, b2, b3, cvec);

  const size_t lds_bytes = (size_t)16 * HID * sizeof(float) + 256 * sizeof(int);
  hipFuncSetAttribute((const void*)mlp_argmax_kernel,
                      hipFuncAttributeMaxDynamicSharedMemorySize, (int)lds_bytes);
  mlp_argmax_kernel<<<B_ROWS / 16, 256, lds_bytes, stream>>>(x, W1, b1, MT, cvec, out);
}